// GlobalAttention_27668179321048
// MI455X (gfx1250) — compile-verified
//
#include <hip/hip_runtime.h>

typedef float v2f __attribute__((ext_vector_type(2)));
typedef float v8f __attribute__((ext_vector_type(8)));

#define BB  32
#define NN  1024
#define FIN 128
#define DD  128

// ws layout (float offsets)
#define WS_ASUM  0      // 32*128
#define WS_WVEC  4096   // 32*128
#define WS_CTERM 8192   // 32
#define WS_U     8224   // 32*128
#define WS_SSUM  12320  // 32

// ---------------- Kernel 1: Asum[b,d] = sum_n aq[b,n,d] ----------------
__global__ __launch_bounds__(128) void k1_colsum(const float* __restrict__ aq,
                                                 float* __restrict__ ws) {
    int b = blockIdx.x, d = threadIdx.x;
    const float* p = aq + (size_t)b * NN * FIN + d;
    float s = 0.f;
#pragma unroll 8
    for (int n = 0; n < NN; ++n) s += p[n * FIN];
    ws[WS_ASUM + b * FIN + d] = s;
}

// ---------------- Kernel 2: WMMA projections ----------------
// Ksum[32,128] = Asum @ Wk + N*bk   (kept in LDS)
// wvec[32,128] = Ksum @ Wq^T        (to ws)
// cterm[b]     = bq . Ksum[b]       (to ws)
__global__ __launch_bounds__(512) void k2_wmma_proj(const float* __restrict__ Wq,
                                                    const float* __restrict__ bq,
                                                    const float* __restrict__ Wk,
                                                    const float* __restrict__ bk,
                                                    float* __restrict__ ws) {
    __shared__ float Ksum[32 * DD];
    int tid = threadIdx.x;
    int lane = tid & 31;
    int w = tid >> 5;            // 16 waves
    int mt = w >> 3;             // 0..1  (M tile, rows = batches)
    int nt = w & 7;              // 0..7  (N tile)
    int lhalf = lane >> 4;       // 0/1
    int lmod = lane & 15;
    int col = nt * 16 + lmod;
    int arow = mt * 16 + lmod;

    const float* Asum = ws + WS_ASUM;

    // Stage A: Ksum = Asum @ Wk + 1024*bk
    v8f acc;
    float bias = 1024.0f * bk[col];
#pragma unroll
    for (int r = 0; r < 8; ++r) acc[r] = bias;
    for (int k0 = 0; k0 < FIN; k0 += 4) {
        int kb = k0 + 2 * lhalf;
        v2f a;  a.x = Asum[arow * FIN + kb];  a.y = Asum[arow * FIN + kb + 1];
        v2f bm; bm.x = Wk[kb * DD + col];     bm.y = Wk[(kb + 1) * DD + col];
        acc = __builtin_amdgcn_wmma_f32_16x16x4_f32(false, a, false, bm,
                                                    (short)0, acc, false, false);
    }
#pragma unroll
    for (int r = 0; r < 8; ++r) {
        int row = mt * 16 + r + 8 * lhalf;
        Ksum[row * DD + col] = acc[r];
    }
    __syncthreads();

    // Stage B: wvec = Ksum @ Wq^T   (B[k,n] = Wq[n,k])
    v8f acc2;
#pragma unroll
    for (int r = 0; r < 8; ++r) acc2[r] = 0.f;
    for (int k0 = 0; k0 < DD; k0 += 4) {
        int kb = k0 + 2 * lhalf;
        v2f a;  a.x = Ksum[arow * DD + kb];   a.y = Ksum[arow * DD + kb + 1];
        v2f bm; bm.x = Wq[col * DD + kb];     bm.y = Wq[col * DD + kb + 1];
        acc2 = __builtin_amdgcn_wmma_f32_16x16x4_f32(false, a, false, bm,
                                                     (short)0, acc2, false, false);
    }
    float* wvec = ws + WS_WVEC;
#pragma unroll
    for (int r = 0; r < 8; ++r) {
        int row = mt * 16 + r + 8 * lhalf;
        wvec[row * FIN + col] = acc2[r];
    }
    __syncthreads();

    if (tid < 32) {
        float c = 0.f;
        for (int d2 = 0; d2 < DD; ++d2) c += bq[d2] * Ksum[tid * DD + d2];
        ws[WS_CTERM + tid] = c;
    }
}

// ---------------- Kernel 3: scores, softmax, weighted reduction ----------------
__global__ __launch_bounds__(256) void k3_attn(const float* __restrict__ aq,
                                               const float* __restrict__ mask,
                                               float* __restrict__ ws,
                                               float* __restrict__ out) {
    __shared__ float sc[NN];     // scores, then weights
    __shared__ float red[256];
    int b = blockIdx.x;
    int tid = threadIdx.x;
    int lane = tid & 31;
    int w = tid >> 5;            // 8 waves
    const float dk = 0.08838834764831845f;   // 128^-0.5

    const float* wvec = ws + WS_WVEC + b * FIN;
    float cterm = ws[WS_CTERM + b];
    float4 wv = *(const float4*)(wvec + lane * 4);
    const float* aqb = aq + (size_t)b * NN * FIN;

    // Pass 1: score[n] = m*dk*(aq[n].wvec + cterm) + (1-m)*(-1e9)
    for (int j = 0; j < NN / 8; ++j) {
        int n = w + 8 * j;
        float4 a = *(const float4*)(aqb + n * FIN + lane * 4);
        float r = a.x * wv.x + a.y * wv.y + a.z * wv.z + a.w * wv.w;
#pragma unroll
        for (int off = 16; off >= 1; off >>= 1) r += __shfl_xor(r, off, 32);
        if (lane == 0) {
            float m = mask[b * NN + n];
            sc[n] = m * dk * (r + cterm) + (1.0f - m) * (-1e9f);
        }
    }
    __syncthreads();

    // max reduce
    float lm = -3.0e38f;
#pragma unroll
    for (int j = 0; j < 4; ++j) lm = fmaxf(lm, sc[tid + 256 * j]);
    red[tid] = lm; __syncthreads();
    for (int s2 = 128; s2 > 0; s2 >>= 1) {
        if (tid < s2) red[tid] = fmaxf(red[tid], red[tid + s2]);
        __syncthreads();
    }
    float mx = red[0]; __syncthreads();

    // exp + sums (Z = sum e, S = sum m*e)
    float ev[4], mv[4];
    float ls = 0.f, lms = 0.f;
#pragma unroll
    for (int j = 0; j < 4; ++j) {
        int n = tid + 256 * j;
        float e = __expf(sc[n] - mx);
        float m = mask[b * NN + n];
        ev[j] = e; mv[j] = m;
        ls += e; lms += m * e;
    }
    red[tid] = ls; __syncthreads();
    for (int s2 = 128; s2 > 0; s2 >>= 1) {
        if (tid < s2) red[tid] += red[tid + s2];
        __syncthreads();
    }
    float Z = red[0]; __syncthreads();
    red[tid] = lms; __syncthreads();
    for (int s2 = 128; s2 > 0; s2 >>= 1) {
        if (tid < s2) red[tid] += red[tid + s2];
        __syncthreads();
    }
    float S = red[0]; __syncthreads();

    float invZ = 1.0f / Z;
#pragma unroll
    for (int j = 0; j < 4; ++j) {
        int n = tid + 256 * j;
        float at = ev[j] * invZ;
        out[b * NN + n] = at;        // attn output
        sc[n] = mv[j] * at;          // weight w = mask*attn
    }
    if (tid == 0) ws[WS_SSUM + b] = S * invZ;
    __syncthreads();

    // U[b,d] = sum_n w[n]*aq[b,n,d]
    int d = tid & 127, half = tid >> 7;
    float u = 0.f;
    for (int n = half; n < NN; n += 2) u += sc[n] * aqb[n * FIN + d];
    red[tid] = u; __syncthreads();
    if (tid < 128) ws[WS_U + b * FIN + tid] = red[tid] + red[tid + 128];
}

// ---------------- Kernel 4: context = U @ Wv + ssum (x) bv ----------------
__global__ __launch_bounds__(512) void k4_context(const float* __restrict__ Wv,
                                                  const float* __restrict__ bv,
                                                  const float* __restrict__ ws,
                                                  float* __restrict__ out) {
    int tid = threadIdx.x;
    int lane = tid & 31;
    int w = tid >> 5;
    int mt = w >> 3, nt = w & 7;
    int lhalf = lane >> 4, lmod = lane & 15;
    int col = nt * 16 + lmod;
    int arow = mt * 16 + lmod;
    const float* U = ws + WS_U;

    v8f acc;
    float bvc = bv[col];
#pragma unroll
    for (int r = 0; r < 8; ++r) {
        int row = mt * 16 + r + 8 * lhalf;
        acc[r] = ws[WS_SSUM + row] * bvc;
    }
    for (int k0 = 0; k0 < FIN; k0 += 4) {
        int kb = k0 + 2 * lhalf;
        v2f a;  a.x = U[arow * FIN + kb];   a.y = U[arow * FIN + kb + 1];
        v2f bm; bm.x = Wv[kb * DD + col];   bm.y = Wv[(kb + 1) * DD + col];
        acc = __builtin_amdgcn_wmma_f32_16x16x4_f32(false, a, false, bm,
                                                    (short)0, acc, false, false);
    }
    float* ctx = out + BB * NN;
#pragma unroll
    for (int r = 0; r < 8; ++r) {
        int row = mt * 16 + r + 8 * lhalf;
        ctx[row * DD + col] = acc[r];
    }
}

extern "C" void kernel_launch(void* const* d_in, const int* in_sizes, int n_in,
                              void* d_out, int out_size, void* d_ws, size_t ws_size,
                              hipStream_t stream) {
    const float* aq   = (const float*)d_in[0];
    const float* mask = (const float*)d_in[1];
    const float* Wq   = (const float*)d_in[2];
    const float* bq   = (const float*)d_in[3];
    const float* Wk   = (const float*)d_in[4];
    const float* bk   = (const float*)d_in[5];
    const float* Wv   = (const float*)d_in[6];
    const float* bv   = (const float*)d_in[7];
    float* out = (float*)d_out;
    float* ws  = (float*)d_ws;

    k1_colsum<<<BB, 128, 0, stream>>>(aq, ws);
    k2_wmma_proj<<<1, 512, 0, stream>>>(Wq, bq, Wk, bk, ws);
    k3_attn<<<BB, 256, 0, stream>>>(aq, mask, ws, out);
    k4_context<<<1, 512, 0, stream>>>(Wv, bv, ws, out);
}